// ImageProjector_47124381171886
// MI455X (gfx1250) — compile-verified
//
#include <hip/hip_runtime.h>
#include <math.h>

// MI455X / gfx1250, wave32. Memory-bound workload (~180MB moved -> ~8us floor
// at 23.3 TB/s). Three passes:
//  1) proj_kernel : WMMA f32 16x16x4 does both 4x4 projective transforms for
//     16 cells/wave; writes img_points, cells, probs (sigmoid), visible flag.
//  2) hmax_kernel : horizontal 9-max; tile staged via GLOBAL_LOAD_ASYNC_TO_LDS
//     (ASYNCcnt path), OOB halo filled with -inf by EXEC-masked ds stores.
//  3) vmax_peak_kernel : vertical 9-max + NMS/threshold -> peak mask.

typedef __attribute__((ext_vector_type(2))) float v2f;
typedef __attribute__((ext_vector_type(8))) float v8f;

#define B_  16
#define N_  512
#define NN_ (N_ * N_)
#define H_  384
#define W_  1280

// rcp + one Newton-Raphson step: <=0.5ulp, avoids the full v_div_* sequence.
__device__ __forceinline__ float fast_rcp(float d) {
  float r = __builtin_amdgcn_rcpf(d);
  r = r * fmaf(-d, r, 2.0f);
  return r;
}

__global__ __launch_bounds__(256) void proj_kernel(
    const float* __restrict__ logits,          // (B,1,N,N)
    const float* __restrict__ height,          // (B,)
    const unsigned char* __restrict__ mask,    // (B,1,H,W) bool (1 byte)
    const float* __restrict__ Kmat,            // (B,4,4)
    const float* __restrict__ Tmat,            // (B,4,4)
    float* __restrict__ img_points,            // (B,M,2) -> (y,x)
    float* __restrict__ probs_out,             // (B,M)
    float* __restrict__ cells_out,             // (B,M,3)
    unsigned char* __restrict__ vis_ws)        // (B,M)
{
  const int b    = blockIdx.y;
  const int lane = threadIdx.x & 31;
  const int wave = threadIdx.x >> 5;
  const int half = lane >> 4;     // 0: lanes 0-15 (K=0/1), 1: lanes 16-31 (K=2/3)
  const int r    = lane & 15;     // A row / cell slot
  const int m    = (blockIdx.x * 8 + wave) * 16 + r;   // cell index (dup per half)
  const int x    = m & (N_ - 1);
  const int z    = m >> 9;

  // ---- P = K * [T[:3]; appended 1] : rows 0..2 only (uniform -> scalar loads)
  const float* Kb = Kmat + b * 16;
  const float* Tb = Tmat + b * 16;
  float P[3][4];
  #pragma unroll
  for (int i = 0; i < 3; ++i) {
    #pragma unroll
    for (int j = 0; j < 4; ++j) {
      float s = 0.f;
      #pragma unroll
      for (int k = 0; k < 3; ++k) s += Kb[i * 4 + k] * Tb[k * 4 + j];
      if (j == 3) s += Kb[i * 4 + 3];     // homogeneous 1 re-appended after T
      P[i][j] = s;
    }
  }

  // A (16x4 f32): row r = P[r][*] for r<3, rest zero.
  // Layout (ISA 7.12.2): lanes0-15 V0=K0,V1=K1 ; lanes16-31 V0=K2,V1=K3.
  v2f a;
  a[0] = (r < 3) ? (half ? P[r][2] : P[r][0]) : 0.f;
  a[1] = (r < 3) ? (half ? P[r][3] : P[r][1]) : 0.f;

  const float y    = height[b];
  const bool  left = x < (N_ >> 1);

  const float sx = (float)x + (left ? 2.f : -1.f);   // sampling offset x
  const float sz = (float)z + 1.f;                   // sampling offset z
  const float cx = (float)x + (left ? 1.f : 0.f);    // proj_left_offset
  const float cz = (float)z;

  // B (4x16 f32): col n = cell n's [x,y,z,1]. V0 = K0|K2, V1 = K1|K3.
  v2f bs, bc;
  bs[0] = half ? sz : sx;  bs[1] = half ? 1.f : y;
  bc[0] = half ? cz : cx;  bc[1] = half ? 1.f : y;

  v8f zero = {};
  v8f ds = __builtin_amdgcn_wmma_f32_16x16x4_f32(false, a, false, bs,
                                                 (short)0, zero, false, false);
  v8f dc = __builtin_amdgcn_wmma_f32_16x16x4_f32(false, a, false, bc,
                                                 (short)0, zero, false, false);

  // D layout: acc VGPR i, lanes 0-15 hold row M=i -> px=ds[0], py=ds[1], pw=ds[2]
  if (lane < 16) {
    // --- visibility via sampling-point projection
    float iw  = fast_rcp(ds[2] + 1e-7f);
    float spx = rintf(ds[0] * iw);                 // round-half-even == jnp.round
    float spy = rintf(ds[1] * iw);
    bool in_range = (spx >= 0.f) & (spx <= (float)(W_ - 1)) &
                    (spy >= 0.f) & (spy <= (float)(H_ - 1));
    int pxc = (int)fminf(fmaxf(spx, 0.f), (float)(W_ - 1));
    int pyc = (int)fminf(fmaxf(spy, 0.f), (float)(H_ - 1));
    unsigned char s = mask[(size_t)b * (H_ * W_) + pyc * W_ + pxc];

    // --- normalized image point from cell projection
    float iw2 = fast_rcp(dc[2] + 1e-7f);
    float xn = fminf(fmaxf(dc[0] * iw2 * (1.f / (float)(W_ - 1)), 0.f), 1.f);
    float yn = fminf(fmaxf(dc[1] * iw2 * (1.f / (float)(H_ - 1)), 0.f), 1.f);

    const size_t o = (size_t)b * NN_ + m;
    float lg = logits[o];
    float p  = fast_rcp(1.f + __expf(-lg));        // sigmoid, no v_div_* chain
    probs_out[o] = p;                              // also pooling input
    vis_ws[o]    = (in_range && s) ? 1 : 0;
    img_points[o * 2 + 0] = yn;
    img_points[o * 2 + 1] = xn;
    cells_out[o * 3 + 0] = cx;
    cells_out[o * 3 + 1] = y;
    cells_out[o * 3 + 2] = cz;
  }
}

// Horizontal 9-wide max: one block = 256 contiguous x of one row. Tile staged
// global->LDS via the CDNA5 async path (no VGPR round trip, ASYNCcnt tracked).
__global__ __launch_bounds__(256) void hmax_kernel(const float* __restrict__ prob,
                                                   float* __restrict__ hmax)
{
  const int b   = blockIdx.y;
  const int z   = blockIdx.x >> 1;       // 2 segments per 512-wide row
  const int seg = blockIdx.x & 1;
  const int tid = threadIdx.x;
  __shared__ float srow[264];            // 256 + 2*4 halo

  const float* row = prob + (size_t)b * NN_ + (size_t)z * N_;

  // slot 0..255 (thread tid -> srow[tid], global x = seg*256 - 4 + tid)
  {
    int g = seg * 256 - 4 + tid;
    unsigned ldsoff = (unsigned)(uintptr_t)(&srow[tid]);   // low32 == LDS offset
    if (g >= 0 && g < N_) {
      int voff = g * 4;
      asm volatile("global_load_async_to_lds_b32 %0, %1, %2"
                   :: "v"(ldsoff), "v"(voff), "s"(row) : "memory");
    } else {
      srow[tid] = -INFINITY;
    }
  }
  // slots 256..263 (tid < 8, global x = seg*256 + 252 + tid)
  if (tid < 8) {
    int g = seg * 256 + 252 + tid;
    unsigned ldsoff = (unsigned)(uintptr_t)(&srow[256 + tid]);
    if (g < N_) {
      int voff = g * 4;
      asm volatile("global_load_async_to_lds_b32 %0, %1, %2"
                   :: "v"(ldsoff), "v"(voff), "s"(row) : "memory");
    } else {
      srow[256 + tid] = -INFINITY;
    }
  }
  asm volatile("s_wait_asynccnt 0x0" ::: "memory");  // my transfers landed
  __syncthreads();                                   // everyone's landed

  float mx = srow[tid];
  #pragma unroll
  for (int k = 1; k < 9; ++k) mx = fmaxf(mx, srow[tid + k]);
  hmax[(size_t)b * NN_ + (size_t)z * N_ + seg * 256 + tid] = mx;
}

// Vertical 9-wide max over hmax + NMS/threshold. Column reads are coalesced
// across x; hmax (16.8MB) sits in the 192MB L2 for the 9x reuse.
__global__ __launch_bounds__(256) void vmax_peak_kernel(
    const float* __restrict__ hmax,
    const float* __restrict__ prob,
    const unsigned char* __restrict__ vis,
    float* __restrict__ peak)
{
  const int b = blockIdx.y;
  const int m = blockIdx.x * 256 + threadIdx.x;
  const int z = m >> 9;
  const int x = m & (N_ - 1);

  const float* base = hmax + (size_t)b * NN_;
  float mx = -INFINITY;
  #pragma unroll
  for (int dz = -4; dz <= 4; ++dz) {
    int zz = z + dz;
    if (zz >= 0 && zz < N_) mx = fmaxf(mx, base[zz * N_ + x]);
  }
  const size_t o = (size_t)b * NN_ + m;
  float p = prob[o];
  peak[o] = (vis[o] && (mx == p) && (p > 0.5f)) ? 1.f : 0.f;
}

extern "C" void kernel_launch(void* const* d_in, const int* in_sizes, int n_in,
                              void* d_out, int out_size, void* d_ws, size_t ws_size,
                              hipStream_t stream) {
  (void)in_sizes; (void)n_in; (void)out_size; (void)ws_size;
  const float*         logits = (const float*)d_in[0];
  const float*         height = (const float*)d_in[1];
  const unsigned char* mask   = (const unsigned char*)d_in[2];  // bool mask, 1B/elt
  const float*         Kmat   = (const float*)d_in[3];
  const float*         Tmat   = (const float*)d_in[4];

  float* img   = (float*)d_out;                        // B*M*2
  float* probs = img   + (size_t)B_ * NN_ * 2;         // B*M   (also intermediate)
  float* cells = probs + (size_t)B_ * NN_;             // B*M*3
  float* peak  = cells + (size_t)B_ * NN_ * 3;         // B*M

  float*         hmax = (float*)d_ws;                  // B*M floats
  unsigned char* vis  = (unsigned char*)(hmax + (size_t)B_ * NN_);

  proj_kernel<<<dim3(NN_ / 128, B_), 256, 0, stream>>>(
      logits, height, mask, Kmat, Tmat, img, probs, cells, vis);
  hmax_kernel<<<dim3(N_ * 2, B_), 256, 0, stream>>>(probs, hmax);
  vmax_peak_kernel<<<dim3(NN_ / 256, B_), 256, 0, stream>>>(hmax, probs, vis, peak);
}